// SAGE_74045236183291
// MI455X (gfx1250) — compile-verified
//
#include <hip/hip_runtime.h>
#include <hip/hip_bf16.h>

#define N_NODES 10000
#define FEAT_D  128
#define TWO_D   256

typedef __attribute__((ext_vector_type(2))) float v2f;
typedef __attribute__((ext_vector_type(8))) float v8f;

// Native CDNA5 fp32 global atomic add, no return (STOREcnt path) — avoids any
// CAS-loop fallback the compiler might pick for atomicAdd(float*).
__device__ __forceinline__ void gadd_f32(float* p, float v) {
    asm volatile("global_atomic_add_f32 %0, %1, off" :: "v"(p), "v"(v) : "memory");
}

__global__ void sage_zero_kernel(float* __restrict__ buf, int n) {
    int i = blockIdx.x * blockDim.x + threadIdx.x;
    const int stride = gridDim.x * blockDim.x;
    for (; i < n; i += stride) buf[i] = 0.0f;
}

// One wave per edge: lanes cooperatively move 128 floats (float4 per lane).
__global__ void sage_scatter_kernel(const float* __restrict__ x,
                                    const int*   __restrict__ ei,   // [2, E]
                                    float* __restrict__ nsum,       // [N, 128]
                                    float* __restrict__ cnt,        // [N]
                                    int E) {
    const int lane = threadIdx.x & 31;
    const int wavesPerBlock = blockDim.x >> 5;
    int w = blockIdx.x * wavesPerBlock + (threadIdx.x >> 5);
    const int nw = gridDim.x * wavesPerBlock;
    for (int e = w; e < E; e += nw) {
        const int src = ei[e];          // row (source of gathered feature)
        const int dst = ei[E + e];      // col (destination of scatter-sum)
        const float4 v = ((const float4*)(x + (size_t)src * FEAT_D))[lane];
        float* d = nsum + (size_t)dst * FEAT_D + lane * 4;
        gadd_f32(d + 0, v.x);
        gadd_f32(d + 1, v.y);
        gadd_f32(d + 2, v.z);
        gadd_f32(d + 3, v.w);
        if (lane == 0) gadd_f32(cnt + dst, 1.0f);
    }
}

// out[m,n] = sum_k h[m,k] * W[n,k] + b[n],  h = concat(x, nsum/max(cnt,1))
// Block = 256 threads = 8 waves. Wave w computes the 16x16 tile at
// rows blockIdx.x*16, cols w*16. K=256 in steps of 4 via fp32 WMMA.
__global__ void sage_gemm_kernel(const float* __restrict__ x,     // [N,128]
                                 const float* __restrict__ nsum,  // [N,128]
                                 const float* __restrict__ cnt,   // [N]
                                 const float* __restrict__ W,     // [128,256] row-major
                                 const float* __restrict__ bias,  // [128]
                                 float* __restrict__ out) {       // [N,128]
    const int lane = threadIdx.x & 31;
    const int wave = threadIdx.x >> 5;
    const int row0 = blockIdx.x * 16;
    const int col0 = wave * 16;

    const int m15  = lane & 15;          // M (A) / N (B) index within tile
    const int kk   = (lane >> 4) << 1;   // lanes 16..31 hold K+2, K+3
    const int mrow = row0 + m15;         // this lane's A row
    const int ncol = col0 + m15;         // this lane's B column

    const float* Ax = x    + (size_t)mrow * FEAT_D;
    const float* Am = nsum + (size_t)mrow * FEAT_D;
    const float* Bw = W    + (size_t)ncol * TWO_D;
    const float inv = 1.0f / fmaxf(cnt[mrow], 1.0f);   // mean-aggr scale

    v8f c = {};
    // K = 0..127 : A from x, B from W[:, 0:128]
    #pragma unroll 8
    for (int k = 0; k < FEAT_D; k += 4) {
        v2f a = *(const v2f*)(Ax + k + kk);
        v2f b = *(const v2f*)(Bw + k + kk);
        c = __builtin_amdgcn_wmma_f32_16x16x4_f32(false, a, false, b,
                                                  (short)0, c, false, false);
    }
    // K = 128..255 : A = mean (nsum * inv), B from W[:, 128:256]
    #pragma unroll 8
    for (int k = 0; k < FEAT_D; k += 4) {
        v2f a = *(const v2f*)(Am + k + kk);
        a.x *= inv; a.y *= inv;
        v2f b = *(const v2f*)(Bw + FEAT_D + k + kk);
        c = __builtin_amdgcn_wmma_f32_16x16x4_f32(false, a, false, b,
                                                  (short)0, c, false, false);
    }

    // D layout: lanes 0-15 -> N=lane, VGPR j -> M=j ; lanes 16-31 -> M=j+8.
    const float bv = bias[ncol];
    const int mbase = row0 + ((lane >> 4) << 3);
    #pragma unroll
    for (int j = 0; j < 8; ++j) {
        out[(size_t)(mbase + j) * FEAT_D + (col0 + m15)] = c[j] + bv;
    }
}

extern "C" void kernel_launch(void* const* d_in, const int* in_sizes, int n_in,
                              void* d_out, int out_size, void* d_ws, size_t ws_size,
                              hipStream_t stream) {
    const float* x    = (const float*)d_in[0];
    const int*   ei   = (const int*)  d_in[1];   // [2, E] int indices
    const float* Wout = (const float*)d_in[2];   // [128, 256]
    const float* bout = (const float*)d_in[3];   // [128]
    float* out = (float*)d_out;

    float* nsum = (float*)d_ws;                          // N*128 floats
    float* cnt  = nsum + (size_t)N_NODES * FEAT_D;       // N floats

    const int E  = in_sizes[1] / 2;
    const int zn = N_NODES * FEAT_D + N_NODES;

    hipLaunchKernelGGL(sage_zero_kernel, dim3(1024), dim3(256), 0, stream,
                       (float*)d_ws, zn);
    hipLaunchKernelGGL(sage_scatter_kernel, dim3(4096), dim3(256), 0, stream,
                       x, ei, nsum, cnt, E);
    hipLaunchKernelGGL(sage_gemm_kernel, dim3(N_NODES / 16), dim3(256), 0, stream,
                       x, nsum, cnt, Wout, bout, out);
}